// CLM_26594437496868
// MI455X (gfx1250) — compile-verified
//
#include <hip/hip_runtime.h>
#include <hip/hip_bf16.h>

typedef float v2f __attribute__((ext_vector_type(2)));
typedef float v8f __attribute__((ext_vector_type(8)));

#define BATCH 4
#define CCH   64
#define HW    4096
#define NEGBIG -1e30f

__device__ __forceinline__ v8f wmma4(v2f a, v2f b, v8f c) {
  // D = A(16x4 f32) * B(4x16 f32) + C(16x16 f32), wave32
  return __builtin_amdgcn_wmma_f32_16x16x4_f32(false, a, false, b, (short)0, c, false, false);
}

// ---------------------------------------------------------------------------
// ex_l[b][n][o] = sum_c exemplar[b][c][n] * W_lin[o][c]
// one wave -> 16(n) x 64(o) tile, K=64 in 16 steps
__global__ void k_exl(const float* __restrict__ ex, const float* __restrict__ Wlin,
                      float* __restrict__ exl) {
  int wid  = (blockIdx.x * blockDim.x + threadIdx.x) >> 5;
  int lane = threadIdx.x & 31;
  int half = lane >> 4, l16 = lane & 15;
  int batch = wid >> 8;
  int n0 = (wid & 255) << 4;
  const float* exb = ex + (size_t)batch * CCH * HW;
  v8f acc[4] = {};
  for (int kk = 0; kk < 16; ++kk) {
    int c0 = kk * 4 + half * 2;
    v2f a;
    a.x = exb[(size_t)c0 * HW + n0 + l16];
    a.y = exb[(size_t)(c0 + 1) * HW + n0 + l16];
    for (int t = 0; t < 4; ++t) {
      v2f bb;
      bb.x = Wlin[(t * 16 + l16) * CCH + c0];
      bb.y = Wlin[(t * 16 + l16) * CCH + c0 + 1];
      acc[t] = wmma4(a, bb, acc[t]);
    }
  }
  for (int t = 0; t < 4; ++t)
    for (int r = 0; r < 8; ++r) {
      int n = n0 + r + half * 8;
      int o = t * 16 + l16;
      exl[((size_t)batch * HW + n) * CCH + o] = acc[t][r];
    }
}

// ---------------------------------------------------------------------------
// A[b][n][m] = sum_k ex_l[b][n][k] * q[b][k][m]
// one wave -> 16(n) x 64(m) tile, K=64  (65536 waves)
__global__ void k_affinity(const float* __restrict__ exl, const float* __restrict__ q,
                           float* __restrict__ A) {
  int wid  = (blockIdx.x * blockDim.x + threadIdx.x) >> 5;
  int lane = threadIdx.x & 31;
  int half = lane >> 4, l16 = lane & 15;
  int batch = wid >> 14;
  int rem = wid & 16383;
  int n0 = (rem >> 6) << 4;
  int m0 = (rem & 63) << 6;
  const float* exlb = exl + (size_t)batch * HW * CCH;
  const float* qb   = q   + (size_t)batch * CCH * HW;
  v8f acc[4] = {};
  for (int kk = 0; kk < 16; ++kk) {
    int k0 = kk * 4 + half * 2;
    v2f a = *(const v2f*)(exlb + (size_t)(n0 + l16) * CCH + k0);
    for (int t = 0; t < 4; ++t) {
      v2f bb;
      bb.x = qb[(size_t)k0 * HW + m0 + t * 16 + l16];
      bb.y = qb[(size_t)(k0 + 1) * HW + m0 + t * 16 + l16];
      acc[t] = wmma4(a, bb, acc[t]);
    }
  }
  float* Ab = A + (size_t)batch * HW * HW;
  for (int t = 0; t < 4; ++t)
    for (int r = 0; r < 8; ++r)
      Ab[(size_t)(n0 + r + half * 8) * HW + m0 + t * 16 + l16] = acc[t][r];
}

// ---------------------------------------------------------------------------
// row stats: rmax[b*HW+n] = max_m A[b][n][m]; rsum = sum exp(A - rmax)
__global__ void k_rowstats(const float* __restrict__ A, float* __restrict__ rmax,
                           float* __restrict__ rsum) {
  int row = blockIdx.x;
  int tid = threadIdx.x;
  const float* Ar = A + (size_t)row * HW;
  float mx = NEGBIG, sm = 0.f;
  for (int m = tid; m < HW; m += 256) {
    float x = Ar[m];
    float nm = fmaxf(mx, x);
    sm = sm * __expf(mx - nm) + __expf(x - nm);
    mx = nm;
  }
  __shared__ float smx[256], ssm[256];
  smx[tid] = mx; ssm[tid] = sm;
  __syncthreads();
  for (int s = 128; s > 0; s >>= 1) {
    if (tid < s) {
      float m1 = smx[tid], m2 = smx[tid + s];
      float nm = fmaxf(m1, m2);
      ssm[tid] = ssm[tid] * __expf(m1 - nm) + ssm[tid + s] * __expf(m2 - nm);
      smx[tid] = nm;
    }
    __syncthreads();
  }
  if (tid == 0) { rmax[row] = smx[0]; rsum[row] = ssm[0]; }
}

// col stats over first index of A, split into 8 m-chunks for parallelism
__global__ void k_colpart(const float* __restrict__ A, float* __restrict__ pmax,
                          float* __restrict__ psum) {
  int bid = blockIdx.x;             // 512 = 4 batch * 16 colgroups * 8 mchunks
  int mc = bid & 7;
  int cg = (bid >> 3) & 15;
  int batch = bid >> 7;
  int n = cg * 256 + threadIdx.x;
  const float* Ab = A + (size_t)batch * HW * HW + n;
  float mx = NEGBIG, sm = 0.f;
  int mend = mc * 512 + 512;
  for (int m = mc * 512; m < mend; ++m) {
    float x = Ab[(size_t)m * HW];
    float nm = fmaxf(mx, x);
    sm = sm * __expf(mx - nm) + __expf(x - nm);
    mx = nm;
  }
  int col = batch * HW + n;
  pmax[(size_t)mc * BATCH * HW + col] = mx;
  psum[(size_t)mc * BATCH * HW + col] = sm;
}

__global__ void k_colmerge(const float* __restrict__ pmax, const float* __restrict__ psum,
                           float* __restrict__ cmax, float* __restrict__ csum) {
  int col = blockIdx.x * blockDim.x + threadIdx.x;   // 16384
  float mx = NEGBIG, sm = 0.f;
  for (int mc = 0; mc < 8; ++mc) {
    float m2 = pmax[(size_t)mc * BATCH * HW + col];
    float s2 = psum[(size_t)mc * BATCH * HW + col];
    float nm = fmaxf(mx, m2);
    sm = sm * __expf(mx - nm) + s2 * __expf(m2 - nm);
    mx = nm;
  }
  cmax[col] = mx; csum[col] = sm;
}

// ---------------------------------------------------------------------------
// split-K attention GEMMs: each block (256 thr = 8 waves) owns one K-chunk of
// 1024 m-values and a 128-wide n range; ex/q A-operand chunks are staged in
// LDS once per block (padded rows -> conflict-free ds_load_b64), shared by
// all 8 waves. Partial sums written per chunk; combined deterministically.
__global__ void k_attend(const float* __restrict__ ex, const float* __restrict__ q,
                         const float* __restrict__ A,
                         const float* __restrict__ rmax, const float* __restrict__ rsum,
                         const float* __restrict__ cmax, const float* __restrict__ csum,
                         float* __restrict__ parts) {
  __shared__ float exs[CCH][66];
  __shared__ float qs[CCH][66];
  int bid = blockIdx.x;             // 512 = 4 batch * 32 ngroups * 4 kchunks
  int kc = bid & 3;
  int g  = (bid >> 2) & 31;
  int batch = bid >> 7;
  int tid = threadIdx.x;
  int w = tid >> 5, lane = tid & 31;
  int half = lane >> 4, l16 = lane & 15;
  int n0 = g * 128 + w * 16;
  int n  = n0 + l16;
  const float* exb = ex + (size_t)batch * CCH * HW;
  const float* qb  = q  + (size_t)batch * CCH * HW;
  const float* Ab  = A  + (size_t)batch * HW * HW;
  float cm = cmax[batch * HW + n], ci = 1.0f / csum[batch * HW + n];
  float rm = rmax[batch * HW + n], ri = 1.0f / rsum[batch * HW + n];
  v8f acc1[4] = {}, acc2[4] = {};
  int mbase = kc * 1024;
  for (int ch = 0; ch < 16; ++ch) {
    int m0 = mbase + ch * 64;
    __syncthreads();                              // previous chunk fully read
    for (int i = tid; i < CCH * 64; i += 256) {   // stage 64x64 chunk of ex,q
      int c = i >> 6, mm = i & 63;
      exs[c][mm] = exb[(size_t)c * HW + m0 + mm];
      qs[c][mm]  = qb[(size_t)c * HW + m0 + mm];
    }
    __syncthreads();
    for (int kk = 0; kk < 16; ++kk) {
      int kl = kk * 4 + half * 2;
      int km = m0 + kl;
      v2f b1, b2;
      b1.x = __expf(Ab[(size_t)km * HW + n] - cm) * ci;        // ColSoftmax(A)
      b1.y = __expf(Ab[(size_t)(km + 1) * HW + n] - cm) * ci;
      v2f araw = *(const v2f*)(Ab + (size_t)n * HW + km);      // RowSoftmax(A)^T
      b2.x = __expf(araw.x - rm) * ri;
      b2.y = __expf(araw.y - rm) * ri;
      for (int t = 0; t < 4; ++t) {
        v2f a1 = *(const v2f*)(&exs[t * 16 + l16][kl]);
        v2f a2 = *(const v2f*)(&qs[t * 16 + l16][kl]);
        acc1[t] = wmma4(a1, b1, acc1[t]);
        acc2[t] = wmma4(a2, b2, acc2[t]);
      }
    }
  }
  float* pb = parts + (size_t)kc * BATCH * CCH * HW + (size_t)batch * CCH * HW;
  for (int t = 0; t < 4; ++t)
    for (int r = 0; r < 8; ++r) {
      int c = t * 16 + r + half * 8;
      pb[(size_t)c * HW + n0 + l16] = acc1[t][r] + acc2[t][r];
    }
}

// deterministic split-K reduction + residual add
__global__ void k_combine(const float* __restrict__ parts, const float* __restrict__ ex,
                          const float* __restrict__ q, float* __restrict__ outx, int ntot) {
  int i = blockIdx.x * blockDim.x + threadIdx.x;
  if (i >= ntot) return;
  const size_t S = (size_t)BATCH * CCH * HW;
  outx[i] = parts[i] + parts[S + i] + parts[2 * S + i] + parts[3 * S + i] + ex[i] + q[i];
}

// ---------------------------------------------------------------------------
// implicit-GEMM 3x3 conv: one wave -> 16(o) x 16(pixel) tile; 9 taps x K=64
// (4096 waves)
__global__ void k_conv(const float* __restrict__ outx, const float* __restrict__ Wc,
                       float* __restrict__ y) {
  int wid  = (blockIdx.x * blockDim.x + threadIdx.x) >> 5;
  int lane = threadIdx.x & 31;
  int half = lane >> 4, l16 = lane & 15;
  int batch = wid >> 10;
  int rem = wid & 1023;
  int p0 = (rem >> 2) << 4;
  int o0 = (rem & 3) << 4;
  int p  = p0 + l16;
  int hh = p >> 6, ww = p & 63;
  const float* xb = outx + (size_t)batch * CCH * HW;
  v8f acc = {};
  for (int dh = -1; dh <= 1; ++dh)
    for (int dw = -1; dw <= 1; ++dw) {
      int h2 = hh + dh, w2 = ww + dw;
      bool valid = ((unsigned)h2 < 64u) && ((unsigned)w2 < 64u);
      int sp = valid ? (h2 * 64 + w2) : 0;        // clamped address, masked value
      float msk = valid ? 1.0f : 0.0f;
      int tap = (dh + 1) * 3 + (dw + 1);
      for (int kk = 0; kk < 16; ++kk) {
        int i0 = kk * 4 + half * 2;
        v2f bb;
        bb.x = xb[(size_t)i0 * HW + sp] * msk;
        bb.y = xb[(size_t)(i0 + 1) * HW + sp] * msk;
        v2f a;
        a.x = Wc[((o0 + l16) * CCH + i0) * 9 + tap];
        a.y = Wc[((o0 + l16) * CCH + i0 + 1) * 9 + tap];
        acc = wmma4(a, bb, acc);
      }
    }
  for (int r = 0; r < 8; ++r) {
    int o = o0 + r + half * 8;
    y[((size_t)batch * CCH + o) * HW + p0 + l16] = acc[r];
  }
}

// ---------------------------------------------------------------------------
__global__ void k_bnstats(const float* __restrict__ y, float* __restrict__ mean,
                          float* __restrict__ istd) {
  int o = blockIdx.x;
  int tid = threadIdx.x;
  float s = 0.f, s2 = 0.f;
  for (int b = 0; b < BATCH; ++b) {
    const float* yb = y + ((size_t)b * CCH + o) * HW;
    for (int p = tid; p < HW; p += 256) {
      float v = yb[p];
      s += v; s2 += v * v;
    }
  }
  __shared__ float sh1[256], sh2[256];
  sh1[tid] = s; sh2[tid] = s2;
  __syncthreads();
  for (int st = 128; st > 0; st >>= 1) {
    if (tid < st) { sh1[tid] += sh1[tid + st]; sh2[tid] += sh2[tid + st]; }
    __syncthreads();
  }
  if (tid == 0) {
    float nrm = 1.0f / (float)(BATCH * HW);
    float m = sh1[0] * nrm;
    float var = sh2[0] * nrm - m * m;
    mean[o] = m;
    istd[o] = rsqrtf(var + 1e-5f);
  }
}

__global__ void k_bnapply(const float* __restrict__ y, const float* __restrict__ mean,
                          const float* __restrict__ istd, const float* __restrict__ gamma,
                          const float* __restrict__ beta, float* __restrict__ out, int ntot) {
  int i = blockIdx.x * blockDim.x + threadIdx.x;
  if (i >= ntot) return;
  int o = (i >> 12) & 63;
  float v = (y[i] - mean[o]) * istd[o] * gamma[o] + beta[o];
  out[i] = v > 0.f ? v : 0.1f * v;
}

// ---------------------------------------------------------------------------
extern "C" void kernel_launch(void* const* d_in, const int* in_sizes, int n_in,
                              void* d_out, int out_size, void* d_ws, size_t ws_size,
                              hipStream_t stream) {
  const float* ex    = (const float*)d_in[0];
  const float* q     = (const float*)d_in[1];
  const float* Wlin  = (const float*)d_in[2];
  const float* Wconv = (const float*)d_in[3];
  const float* gamma = (const float*)d_in[4];
  const float* beta  = (const float*)d_in[5];
  float* out = (float*)d_out;

  float* ws = (float*)d_ws;
  const size_t nA    = (size_t)BATCH * HW * HW;     // 268 MB
  const size_t nExl  = (size_t)BATCH * HW * CCH;
  const size_t nStat = (size_t)BATCH * HW;
  const size_t nX    = (size_t)BATCH * CCH * HW;
  float* A     = ws;
  float* exl   = A + nA;
  float* rmax  = exl + nExl;
  float* rsum  = rmax + nStat;
  float* cmax  = rsum + nStat;
  float* csum  = cmax + nStat;
  float* pmax  = csum + nStat;        // 8 * nStat
  float* psum  = pmax + 8 * nStat;    // 8 * nStat
  float* parts = psum + 8 * nStat;    // 4 * nX
  float* outx  = parts + 4 * nX;
  float* yb    = outx + nX;
  float* mean  = yb + nX;
  float* istd  = mean + CCH;

  dim3 blk(256);
  k_exl     <<<dim3(128),        blk, 0, stream>>>(ex, Wlin, exl);
  k_affinity<<<dim3(8192),       blk, 0, stream>>>(exl, q, A);
  k_rowstats<<<dim3(BATCH * HW), blk, 0, stream>>>(A, rmax, rsum);
  k_colpart <<<dim3(512),        blk, 0, stream>>>(A, pmax, psum);
  k_colmerge<<<dim3(64),         blk, 0, stream>>>(pmax, psum, cmax, csum);
  k_attend  <<<dim3(512),        blk, 0, stream>>>(ex, q, A, rmax, rsum, cmax, csum, parts);
  k_combine <<<dim3((int)((nX + 255) / 256)), blk, 0, stream>>>(parts, ex, q, outx, (int)nX);
  k_conv    <<<dim3(512),        blk, 0, stream>>>(outx, Wconv, yb);
  k_bnstats <<<dim3(CCH),        blk, 0, stream>>>(yb, mean, istd);
  k_bnapply <<<dim3((int)((nX + 255) / 256)), blk, 0, stream>>>(yb, mean, istd, gamma, beta,
                                                                out, (int)nX);
}